// BilinearPairedLayer_67087389164341
// MI455X (gfx1250) — compile-verified
//
#include <hip/hip_runtime.h>

typedef __attribute__((ext_vector_type(2))) float v2f;
typedef __attribute__((ext_vector_type(4))) float v4f;
typedef __attribute__((ext_vector_type(8))) float v8f;

#define NB   4
#define NN   1024
#define NIN  64
#define NHID 32
#define NOUT 8
#define HH   96   // CONTEXT * NHID

// ---------------------------------------------------------------------------
// Kernel 1: h[b,n,k] = relu(x[b,n,:] @ W_lin[:,k] + b_lin[k])   [4096 x 32]
// ---------------------------------------------------------------------------
__global__ __launch_bounds__(256) void k_lin_relu(
    const float* __restrict__ x, const float* __restrict__ W,
    const float* __restrict__ bl, float* __restrict__ h) {
  int tid = blockIdx.x * 256 + threadIdx.x;   // B*N*NHID = 131072 threads
  int k  = tid & (NHID - 1);
  int bn = tid >> 5;
  const float* xr = x + (size_t)bn * NIN;
  float s = bl[k];
#pragma unroll 8
  for (int i = 0; i < NIN; ++i) s = fmaf(xr[i], W[i * NHID + k], s);
  h[tid] = fmaxf(s, 0.0f);
}

// ---------------------------------------------------------------------------
// Kernel 2: v[b,n,c*32+k] = h[b, n + c - 1, k]  (zero padded)   [4096 x 96]
// ---------------------------------------------------------------------------
__global__ __launch_bounds__(256) void k_shift_concat(
    const float* __restrict__ h, float* __restrict__ v) {
  int tid = blockIdx.x * 256 + threadIdx.x;   // B*N*HH = 393216 threads
  int g  = tid % HH;
  int bn = tid / HH;
  int b  = bn >> 10;
  int n  = bn & (NN - 1);
  int c  = g >> 5;           // 0,1,2  -> shift d = c-1
  int k  = g & 31;
  int sn = n + c - 1;
  float val = 0.0f;
  if (sn >= 0 && sn < NN) val = h[(((size_t)b << 10) + sn) * NHID + k];
  v[tid] = val;
}

// ---------------------------------------------------------------------------
// Kernel 3 (WMMA): t[bi, o*96+g] = sum_k v[bi,k] * W_bil[o,k,g]
// GEMM: C[4096, 768] = v[4096,96] x B[96,768], B[k, o*96+g] = W_bil[o,k,g].
// One wave = 16 bi-rows x one o block; sweeps 6 g-tiles of 16, A in registers.
// ---------------------------------------------------------------------------
__global__ __launch_bounds__(256) void k_t_wmma(
    const float* __restrict__ v, const float* __restrict__ Wb,
    float* __restrict__ t) {
  const int lane   = threadIdx.x & 31;
  const int laneLo = lane & 15;
  const bool hi    = lane >= 16;

  int wave = (blockIdx.x * 256 + threadIdx.x) >> 5;  // 2048 waves
  int o   = wave & (NOUT - 1);
  int bi0 = (wave >> 3) * 16;                        // 256 row tiles

  // A fragments from v rows: lane<16 K={4s,4s+1}; lane>=16 K={4s+2,4s+3}
  const float* vA = v + (size_t)(bi0 + laneLo) * HH + (hi ? 2 : 0);
  v2f a[24];
#pragma unroll
  for (int s = 0; s < 24; ++s) a[s] = *(const v2f*)(vA + s * 4);

  // B base for this o block; hi lanes offset by 2 K-rows (2*HH floats)
  const float* wB = Wb + (size_t)o * HH * HH + (hi ? 2 * HH : 0) + laneLo;

#pragma unroll
  for (int gt = 0; gt < 6; ++gt) {
    const int g0 = gt * 16;
    v8f acc = {};
#pragma unroll
    for (int s = 0; s < 24; ++s) {
      const float* p = wB + (size_t)(4 * s) * HH + g0;
      v2f bf = { p[0], p[HH] };    // K and K+1 rows of W_bil[o]
      acc = __builtin_amdgcn_wmma_f32_16x16x4_f32(
          false, a[s], false, bf, (short)0, acc, false, false);
    }
    // D row m = bi0 + r (+8 for hi lanes); col = o*96 + g0 + laneLo
    float* tp = t + (size_t)(bi0 + (hi ? 8 : 0)) * (NOUT * HH)
                  + o * HH + g0 + laneLo;
#pragma unroll
    for (int r = 0; r < 8; ++r)
      tp[(size_t)r * (NOUT * HH)] = acc[r];
  }
}

// ---------------------------------------------------------------------------
// Kernel 4 (main): per batch, C[m=i*8+o, j] = sum_g t[b,m,g] * v[b,j,g] + bias
// WMMA f32 16x16x4, K=96 in 24 steps. One wave = 16 M-rows x 4 j-tiles.
// 16 M-rows = 2 i values x 8 o  ->  out[b,i,j0:j0+16,0:8] contiguous stores.
// ---------------------------------------------------------------------------
__global__ __launch_bounds__(256) void k_bilinear_out(
    const float* __restrict__ t, const float* __restrict__ v,
    const float* __restrict__ bb_g, float* __restrict__ out) {
  const int lane   = threadIdx.x & 31;
  const int laneLo = lane & 15;
  const bool hi    = lane >= 16;

  int wave = (blockIdx.x * 256 + threadIdx.x) >> 5;  // 32768 waves total
  int jg = wave & 15;            // 16 groups of 4 j-tiles (j covers 1024)
  int mt = (wave >> 4) & 511;    // 512 m-tiles (m covers 8192)
  int b  = wave >> 13;           // 4 batches

  const int m0 = mt * 16;
  const int i0 = m0 >> 3;        // rows m0..m0+7 -> i0 ; m0+8..m0+15 -> i0+1

  // Bias (uniform -> scalar loads)
  float bb[NOUT];
#pragma unroll
  for (int r = 0; r < NOUT; ++r) bb[r] = bb_g[r];

  // A fragments: lane<16 holds K={4s,4s+1}; lane>=16 holds K={4s+2,4s+3}
  const float* tA = t + ((size_t)b * 8192 + m0 + laneLo) * HH + (hi ? 2 : 0);
  v2f a[24];
#pragma unroll
  for (int s = 0; s < 24; ++s) a[s] = *(const v2f*)(tA + s * 4);

  const float* vBb = v + (size_t)b * NN * HH + (hi ? 2 : 0);
  const int i = i0 + (hi ? 1 : 0);

#pragma unroll
  for (int tj = 0; tj < 4; ++tj) {
    const int j0 = (jg * 4 + tj) * 16;
    const float* vB = vBb + (size_t)(j0 + laneLo) * HH;

    v8f acc = {};
#pragma unroll
    for (int s = 0; s < 24; ++s) {
      v2f bf = *(const v2f*)(vB + s * 4);
      acc = __builtin_amdgcn_wmma_f32_16x16x4_f32(
          /*neg_a=*/false, a[s], /*neg_b=*/false, bf,
          /*c_mod=*/(short)0, acc, /*reuse_a=*/false, /*reuse_b=*/false);
    }

    // out[b, i(lane half), j0+laneLo, 0:8] : 8 contiguous floats per lane
    const int j = j0 + laneLo;
    float* op = out + (((size_t)b * NN + i) * NN + j) * NOUT;
    v4f lo = { acc[0] + bb[0], acc[1] + bb[1], acc[2] + bb[2], acc[3] + bb[3] };
    v4f hi4 = { acc[4] + bb[4], acc[5] + bb[5], acc[6] + bb[6], acc[7] + bb[7] };
    __builtin_nontemporal_store(lo,  (v4f*)op);
    __builtin_nontemporal_store(hi4, (v4f*)(op + 4));
  }
}

// ---------------------------------------------------------------------------
extern "C" void kernel_launch(void* const* d_in, const int* in_sizes, int n_in,
                              void* d_out, int out_size, void* d_ws, size_t ws_size,
                              hipStream_t stream) {
  const float* x     = (const float*)d_in[0];   // [4,1024,64]
  const float* W_lin = (const float*)d_in[1];   // [64,32]
  const float* b_lin = (const float*)d_in[2];   // [32]
  const float* W_bil = (const float*)d_in[3];   // [8,96,96]
  const float* b_bil = (const float*)d_in[4];   // [8]
  float* out = (float*)d_out;                   // [4,1024,1024,8]

  char* ws = (char*)d_ws;
  float* h  = (float*)ws;                                   // 131072 f = 0.5 MB
  float* v  = (float*)(ws + (size_t)NB * NN * NHID * 4);    // 393216 f = 1.5 MB
  float* t  = (float*)(ws + (size_t)NB * NN * NHID * 4
                          + (size_t)NB * NN * HH * 4);      // 3145728 f = 12 MB

  k_lin_relu<<<(NB * NN * NHID) / 256, 256, 0, stream>>>(x, W_lin, b_lin, h);
  k_shift_concat<<<(NB * NN * HH) / 256, 256, 0, stream>>>(h, v);
  // 2048 waves = 65536 threads
  k_t_wmma<<<256, 256, 0, stream>>>(v, W_bil, t);
  // 32768 waves = 1,048,576 threads
  k_bilinear_out<<<4096, 256, 0, stream>>>(t, v, b_bil, out);
}